// BilinearInterpolation_1486058684864
// MI455X (gfx1250) — compile-verified
//
#include <hip/hip_runtime.h>

#define IMG_H 512
#define IMG_W 512

typedef float v2f __attribute__((ext_vector_type(2)));
typedef float v8f __attribute__((ext_vector_type(8)));

struct __attribute__((aligned(4))) F3 { float x, y, z; };

// One wave32 per 16x16 output tile. 8 waves (256 threads) per block.
// blocks-per-image = 1024 tiles / 8 waves = 128 -> batch index is
// block-uniform, so theta loads lower to scalar (s_load) reads.
__global__ __launch_bounds__(256) void bilerp_wmma_kernel(
    const float* __restrict__ images,   // [B,512,512,3]
    const float* __restrict__ theta,    // [B,2,3]
    float* __restrict__ out)            // [B,512,512,3]
{
  const int wave = threadIdx.x >> 5;
  const int lane = threadIdx.x & 31;
  const int b    = blockIdx.x >> 7;                    // image index (uniform)
  const int tile = ((blockIdx.x & 127) << 3) + wave;   // 0..1023 within image
  const int ty   = tile >> 5;
  const int tx   = tile & 31;
  const int row0 = ty << 4;
  const int col0 = tx << 4;

  const float* th = theta + b * 6;
  const float t00 = th[0], t01 = th[1], t02 = th[2];
  const float t10 = th[3], t11 = th[4], t12 = th[5];

  const float step = 2.0f / 511.0f;  // jnp.linspace(-1,1,512) step
  const int   m    = lane & 15;      // row index for A, col index for B/D
  const int   hi   = lane >> 4;      // 0: K=0,1 half   1: K=2,3 half (zeroed)

  const float gy = -1.0f + (float)(row0 + m) * step;
  const float gx = -1.0f + (float)(col0 + m) * step;

  // A (16x4): per-lane VGPR0=K(0|2), VGPR1=K(1|3). K>=2 rows are zero.
  v2f ax, ay, bv;
  ax[0] = hi ? 0.0f : t00;
  ax[1] = hi ? 0.0f : fmaf(t01, gy, t02);
  ay[0] = hi ? 0.0f : t10;
  ay[1] = hi ? 0.0f : fmaf(t11, gy, t12);
  // B (4x16): K=0 row = gx(col), K=1 row = 1.0, K=2,3 rows = 0.
  bv[0] = hi ? 0.0f : gx;
  bv[1] = hi ? 0.0f : 1.0f;

  v8f c0 = {};
  // D[r][n] = theta_row0 . [gx(n), gy(r), 1]  for the 16x16 tile
  v8f xs = __builtin_amdgcn_wmma_f32_16x16x4_f32(
      false, ax, false, bv, (short)0, c0, false, false);
  v8f ys = __builtin_amdgcn_wmma_f32_16x16x4_f32(
      false, ay, false, bv, (short)0, c0, false, false);

  const float* img  = images + (size_t)b * (IMG_H * IMG_W * 3);
  float*       outb = out    + (size_t)b * (IMG_H * IMG_W * 3);
  const int    col  = col0 + m;

#pragma unroll
  for (int j = 0; j < 8; ++j) {
    const int row = row0 + j + (hi << 3);  // D layout: lanes16-31 hold M=8+j

    float x = (xs[j] + 1.0f) * (0.5f * IMG_W);
    float y = (ys[j] + 1.0f) * (0.5f * IMG_H);

    float x0f = fminf(fmaxf(floorf(x), 0.0f), (float)(IMG_W - 1));
    float y0f = fminf(fmaxf(floorf(y), 0.0f), (float)(IMG_H - 1));
    int   x0  = (int)x0f;
    int   y0  = (int)y0f;
    int   x1  = x0 + 1 > IMG_W - 1 ? IMG_W - 1 : x0 + 1;
    int   y1  = y0 + 1 > IMG_H - 1 ? IMG_H - 1 : y0 + 1;
    float x1f = (float)x1;
    float y1f = (float)y1;

    float xc = fminf(fmaxf(x, 0.0f), (float)(IMG_W - 1));
    float yc = fminf(fmaxf(y, 0.0f), (float)(IMG_H - 1));

    float wx0 = x1f - xc;   // weight toward x0
    float wx1 = xc - x0f;   // weight toward x1
    float wy0 = y1f - yc;
    float wy1 = yc - y0f;

    float wa = wx0 * wy0;   // (x0,y0)
    float wb = wx0 * wy1;   // (x0,y1)
    float wc = wx1 * wy0;   // (x1,y0)
    float wd = wx1 * wy1;   // (x1,y1)

    const F3 Ia = *(const F3*)(img + (size_t)(y0 * IMG_W + x0) * 3);
    const F3 Ib = *(const F3*)(img + (size_t)(y1 * IMG_W + x0) * 3);
    const F3 Ic = *(const F3*)(img + (size_t)(y0 * IMG_W + x1) * 3);
    const F3 Id = *(const F3*)(img + (size_t)(y1 * IMG_W + x1) * 3);

    F3 r;
    r.x = fmaf(wa, Ia.x, fmaf(wb, Ib.x, fmaf(wc, Ic.x, wd * Id.x)));
    r.y = fmaf(wa, Ia.y, fmaf(wb, Ib.y, fmaf(wc, Ic.y, wd * Id.y)));
    r.z = fmaf(wa, Ia.z, fmaf(wb, Ib.z, fmaf(wc, Ic.z, wd * Id.z)));

    *(F3*)(outb + (size_t)(row * IMG_W + col) * 3) = r;
  }
}

extern "C" void kernel_launch(void* const* d_in, const int* in_sizes, int n_in,
                              void* d_out, int out_size, void* d_ws, size_t ws_size,
                              hipStream_t stream) {
  const float* images = (const float*)d_in[0];
  const float* theta  = (const float*)d_in[1];
  float*       out    = (float*)d_out;

  const int B = in_sizes[1] / 6;     // 32 images
  const int blocks = B * 128;        // 128 blocks/image, 8 waves/block, 1 tile/wave

  bilerp_wmma_kernel<<<blocks, 256, 0, stream>>>(images, theta, out);
}